// qlstm_autoencoder_23622320128523
// MI455X (gfx1250) — compile-verified
//
#include <hip/hip_runtime.h>
#include <hip/hip_bf16.h>

#ifndef __has_builtin
#define __has_builtin(x) 0
#endif

typedef __attribute__((ext_vector_type(2))) float v2f;
typedef __attribute__((ext_vector_type(8))) float v8f;
typedef __attribute__((ext_vector_type(4))) int   v4i;

#define NQ 4
#define TT 512
#define HH 1024
#define BB 16
#define NTHREADS 512          // 16 waves (wave32)
#define KMAXF (HH + 64)       // encoder K = H+F

// ---- dynamic LDS layout (floats) ----
#define GW_F   (NQ * HH * 4)            // 16384: 4 gates x (4 x 1024) weights
#define WY_OFF (GW_F)
#define WY_F   (NQ * KMAXF)             // 4352
#define GB_OFF (WY_OFF + WY_F)          // 20736
#define GB_F   (NQ * HH)                // 4096
#define BY_OFF (GB_OFF + GB_F)          // 24832
#define BUF_F  (BY_OFF + 8)             // 24840 (16B aligned)
#define H_OFF  (2 * BUF_F)              // 49680
#define ZACT_OFF (H_OFF + BB * HH)      // 66064
#define YACC_OFF (ZACT_OFF + 4 * BB * NQ)
#define SMEM_F (YACC_OFF + BB * NQ)     // 66384 floats = 265536 B  (< 320KB/WGP)

#if defined(__HIP_DEVICE_COMPILE__) && __has_builtin(__builtin_amdgcn_global_load_async_to_lds_b128)
#define HAVE_ASYNC 1
#else
#define HAVE_ASYNC 0
#endif

#define AS1 __attribute__((address_space(1)))
#define AS3 __attribute__((address_space(3)))

// native v_tanh_f32 on gfx1250
__device__ __forceinline__ float fast_tanh(float x) {
#if defined(__HIP_DEVICE_COMPILE__) && __has_builtin(__builtin_amdgcn_tanhf)
  return __builtin_amdgcn_tanhf(x);
#else
  return tanhf(x);
#endif
}

__device__ __forceinline__ float fast_sigmoid(float x) {
  return __builtin_amdgcn_rcpf(1.f + __expf(-x));   // v_exp_f32 + v_rcp_f32
}

__device__ __forceinline__ void stage16(const float* g, float* l) {
#if HAVE_ASYNC
  float* gg = const_cast<float*>(g);
  __builtin_amdgcn_global_load_async_to_lds_b128((AS1 v4i*)gg, (AS3 v4i*)l, 0, 0);
#else
  float4 v = *(const float4*)g;
  *(float4*)l = v;
#endif
}

__device__ __forceinline__ void async_wait() {
#if HAVE_ASYNC
  asm volatile("s_wait_asynccnt 0" ::: "memory");
#endif
}

__device__ __forceinline__ void stage_block(float* dst, const float* src, int nf, int tid) {
  for (int i = tid * 4; i < nf; i += NTHREADS * 4) stage16(src + i, dst + i);
}

struct StepParams {
  const float* x; int sxb; int sxt; int Fin; int K;
  const float* qw0; const float* qw1; const float* qw2; const float* qw3;
  const float* Wy; const float* by;
  const float* Wf; const float* bf;
  const float* Wi; const float* bi;
  const float* Wu; const float* bu;
  const float* Wo; const float* bo;
  float* hout;
};

__device__ __forceinline__ void stage_all(float* buf, const StepParams& p, int t, int tid) {
  stage_block(buf + 0 * NQ * HH, p.Wf + (size_t)t * NQ * HH, NQ * HH, tid);
  stage_block(buf + 1 * NQ * HH, p.Wi + (size_t)t * NQ * HH, NQ * HH, tid);
  stage_block(buf + 2 * NQ * HH, p.Wu + (size_t)t * NQ * HH, NQ * HH, tid);
  stage_block(buf + 3 * NQ * HH, p.Wo + (size_t)t * NQ * HH, NQ * HH, tid);
  stage_block(buf + WY_OFF, p.Wy + (size_t)t * p.K * NQ, p.K * NQ, tid);
  stage_block(buf + GB_OFF + 0 * HH, p.bf + (size_t)t * HH, HH, tid);
  stage_block(buf + GB_OFF + 1 * HH, p.bi + (size_t)t * HH, HH, tid);
  stage_block(buf + GB_OFF + 2 * HH, p.bu + (size_t)t * HH, HH, tid);
  stage_block(buf + GB_OFF + 3 * HH, p.bo + (size_t)t * HH, HH, tid);
  if (tid == 0) stage16(p.by + (size_t)t * NQ, buf + BY_OFF);
}

// Persistent single-workgroup QLSTM: h in LDS, c in D-layout registers,
// per-step weights double-buffered (async global->LDS), gates via
// v_wmma_f32_16x16x4_f32.
__global__ __launch_bounds__(NTHREADS)
void qlstm_step(StepParams p) {
  extern __shared__ float sm[];
  float* hL   = sm + H_OFF;
  float* zact = sm + ZACT_OFF;
  float* yacc = sm + YACC_OFF;

  const int tid  = threadIdx.x;
  const int lane = tid & 31;
  const int wave = tid >> 5;          // 0..15
  const int hi   = lane >> 4;         // 0/1
  const int lo   = lane & 15;

  for (int i = tid; i < BB * HH; i += NTHREADS) hL[i] = 0.f;

  v8f cst[4];
  #pragma unroll
  for (int j = 0; j < 4; ++j)
    #pragma unroll
    for (int r = 0; r < 8; ++r) cst[j][r] = 0.f;

  stage_all(sm, p, 0, tid);
  async_wait();
  __syncthreads();

  for (int t = 0; t < TT; ++t) {
    float* cb = sm + (t & 1) * BUF_F;
    float* nb = sm + ((t & 1) ^ 1) * BUF_F;
    if (t + 1 < TT) stage_all(nb, p, t + 1, tid);   // prefetch next step into other buffer

    // ---- phase B: y[b][:] = [h, x_t] . Wy[t]  (wave-per-batch-row, shfl reduce) ----
    {
      const int b = wave;                       // each wave owns one batch row
      const float* wy = cb + WY_OFF;            // [k*4 + q]
      const float* xb = p.x + (size_t)b * p.sxb + (size_t)t * p.sxt;
      const int ch = (p.K + 31) >> 5;           // contiguous k-chunk per lane
      const int k0 = lane * ch;
      const int k1 = (k0 + ch < p.K) ? (k0 + ch) : p.K;
      float a0 = 0.f, a1 = 0.f, a2 = 0.f, a3 = 0.f;
      for (int k = k0; k < k1; ++k) {
        float v = (k < HH) ? hL[b * HH + k] : xb[k - HH];
        float4 w = *(const float4*)(wy + k * 4);   // ds_load_b128
        a0 = fmaf(v, w.x, a0); a1 = fmaf(v, w.y, a1);
        a2 = fmaf(v, w.z, a2); a3 = fmaf(v, w.w, a3);
      }
      #pragma unroll
      for (int off = 16; off > 0; off >>= 1) {
        a0 += __shfl_xor(a0, off, 32);
        a1 += __shfl_xor(a1, off, 32);
        a2 += __shfl_xor(a2, off, 32);
        a3 += __shfl_xor(a3, off, 32);
      }
      if (lane == 0) {
        yacc[b * 4 + 0] = a0; yacc[b * 4 + 1] = a1;
        yacc[b * 4 + 2] = a2; yacc[b * 4 + 3] = a3;
      }
    }
    __syncthreads();

    // ---- phase C: 4-qubit circuit expvals + activation -> A matrices ----
    if (tid < 64) {
      const int g = tid >> 4, b = tid & 15;
      const float* qw = (g == 0) ? p.qw0 : (g == 1) ? p.qw1 : (g == 2) ? p.qw2 : p.qw3;
      const float* byt = cb + BY_OFF;
      float c2[4], s2[4];
      #pragma unroll
      for (int q = 0; q < 4; ++q) {
        float th = 0.5f * (yacc[b * 4 + q] + byt[q] + qw[q]);
        float cs = __cosf(th), sn = __sinf(th);
        c2[q] = cs * cs; s2[q] = sn * sn;
      }
      float e0 = 0.f, e1 = 0.f, e2 = 0.f, e3 = 0.f;
      #pragma unroll
      for (int i = 0; i < 16; ++i) {
        int j = i;                               // compose the 4 CNOT-ring perms
        #pragma unroll
        for (int c = 3; c >= 0; --c) {
          int tq = (c + 1) & 3;
          j ^= ((j >> (3 - c)) & 1) << (3 - tq);
        }
        float pr = ((j & 8) ? s2[0] : c2[0]) * ((j & 4) ? s2[1] : c2[1])
                 * ((j & 2) ? s2[2] : c2[2]) * ((j & 1) ? s2[3] : c2[3]);
        e0 += (i & 8) ? -pr : pr;
        e1 += (i & 4) ? -pr : pr;
        e2 += (i & 2) ? -pr : pr;
        e3 += (i & 1) ? -pr : pr;
      }
      float e[4] = {e0, e1, e2, e3};
      #pragma unroll
      for (int q = 0; q < 4; ++q) {
        float z = e[q];
        float a = (g == 2) ? fast_tanh(z) : fast_sigmoid(z);
        zact[g * 64 + b * 4 + q] = a;            // A-matrix source, [g][M][K]
      }
    }
    __syncthreads();

    // ---- phase D: 4 gate WMMAs per H-tile, fused LSTM cell update ----
    v2f a[4];
    #pragma unroll
    for (int g = 0; g < 4; ++g) {               // A 16x4: vgpr r = K=2*hi+r, lane = M
      a[g].x = zact[g * 64 + lo * 4 + 2 * hi + 0];
      a[g].y = zact[g * 64 + lo * 4 + 2 * hi + 1];
    }
    const float* gw = cb;                        // [g*4096 + k*1024 + n]
    const float* gb = cb + GB_OFF;               // [g*1024 + n]
    #pragma unroll
    for (int j = 0; j < 4; ++j) {
      const int n = ((wave + 16 * j) << 4) + lo; // this lane's N column
      v8f d[4];
      #pragma unroll
      for (int g = 0; g < 4; ++g) {
        v2f bv;                                  // B 4x16: vgpr r = K=2*hi+r, lane = N
        bv.x = gw[g * 4096 + (2 * hi + 0) * HH + n];
        bv.y = gw[g * 4096 + (2 * hi + 1) * HH + n];
        float bb = gb[g * 1024 + n];
        v8f cc;
        #pragma unroll
        for (int r = 0; r < 8; ++r) cc[r] = bb;  // bias depends only on N
        d[g] = __builtin_amdgcn_wmma_f32_16x16x4_f32(
                   false, a[g], false, bv, (short)0, cc, false, false);
      }
      v8f c = cst[j];
      #pragma unroll
      for (int r = 0; r < 8; ++r) {              // D layout: M = r + 8*hi, N = n
        float cv = fmaf(d[0][r], c[r], d[1][r] * d[2][r]);  // f*c + i*g
        c[r] = cv;
        hL[(r + 8 * hi) * HH + n] = d[3][r] * fast_tanh(cv); // h = o*tanh(c)
      }
      cst[j] = c;
    }
    async_wait();        // next-step weights landed
    __syncthreads();     // h visible; current buffer free for t+2 staging
  }

  for (int i = tid; i < BB * HH; i += NTHREADS) p.hout[i] = hL[i];
}

// target = h_dec @ Wt + bt   via v_wmma_f32_16x16x4_f32 (16x512x1024)
__global__ __launch_bounds__(256)
void qae_final(const float* __restrict__ h, const float* __restrict__ Wt,
               const float* __restrict__ bt, float* __restrict__ out) {
  const int lane = threadIdx.x & 31;
  const int wave = threadIdx.x >> 5;
  const int tile = blockIdx.x * 8 + wave;        // 32 tiles of 16 columns
  const int hi = lane >> 4, lo = lane & 15;
  const int n = tile * 16 + lo;
  v8f acc;
  float bb = bt[n];
  #pragma unroll
  for (int r = 0; r < 8; ++r) acc[r] = bb;
  for (int k0 = 0; k0 < HH; k0 += 4) {
    v2f a, b;
    a.x = h[lo * HH + k0 + 2 * hi + 0];
    a.y = h[lo * HH + k0 + 2 * hi + 1];
    b.x = Wt[(size_t)(k0 + 2 * hi + 0) * TT + n];
    b.y = Wt[(size_t)(k0 + 2 * hi + 1) * TT + n];
    acc = __builtin_amdgcn_wmma_f32_16x16x4_f32(false, a, false, b, (short)0, acc, false, false);
  }
  #pragma unroll
  for (int r = 0; r < 8; ++r) out[(size_t)(r + 8 * hi) * TT + n] = acc[r];
}

extern "C" void kernel_launch(void* const* d_in, const int* in_sizes, int n_in,
                              void* d_out, int out_size, void* d_ws, size_t ws_size,
                              hipStream_t stream) {
  (void)in_sizes; (void)n_in; (void)out_size; (void)ws_size;
  const float* x    = (const float*)d_in[0];
  const float* qwf  = (const float*)d_in[1];
  const float* qwi  = (const float*)d_in[2];
  const float* qwu  = (const float*)d_in[3];
  const float* qwo  = (const float*)d_in[4];
  const float* Wy_e = (const float*)d_in[5];
  const float* by_e = (const float*)d_in[6];
  const float* Wf_e = (const float*)d_in[7];
  const float* bf_e = (const float*)d_in[8];
  const float* Wi_e = (const float*)d_in[9];
  const float* bi_e = (const float*)d_in[10];
  const float* Wu_e = (const float*)d_in[11];
  const float* bu_e = (const float*)d_in[12];
  const float* Wo_e = (const float*)d_in[13];
  const float* bo_e = (const float*)d_in[14];
  const float* Wy_d = (const float*)d_in[15];
  const float* by_d = (const float*)d_in[16];
  const float* Wf_d = (const float*)d_in[17];
  const float* bf_d = (const float*)d_in[18];
  const float* Wi_d = (const float*)d_in[19];
  const float* bi_d = (const float*)d_in[20];
  const float* Wu_d = (const float*)d_in[21];
  const float* bu_d = (const float*)d_in[22];
  const float* Wo_d = (const float*)d_in[23];
  const float* bo_d = (const float*)d_in[24];
  const float* Wt   = (const float*)d_in[25];
  const float* bt   = (const float*)d_in[26];

  float* henc = (float*)d_ws;
  float* hdec = henc + BB * HH;

  const size_t smem = (size_t)SMEM_F * sizeof(float);
  (void)hipFuncSetAttribute((const void*)qlstm_step,
                            hipFuncAttributeMaxDynamicSharedMemorySize, (int)smem);

  StepParams pe;
  pe.x = x; pe.sxb = TT * 64; pe.sxt = 64; pe.Fin = 64; pe.K = HH + 64;
  pe.qw0 = qwf; pe.qw1 = qwi; pe.qw2 = qwu; pe.qw3 = qwo;
  pe.Wy = Wy_e; pe.by = by_e;
  pe.Wf = Wf_e; pe.bf = bf_e; pe.Wi = Wi_e; pe.bi = bi_e;
  pe.Wu = Wu_e; pe.bu = bu_e; pe.Wo = Wo_e; pe.bo = bo_e;
  pe.hout = henc;
  qlstm_step<<<1, NTHREADS, smem, stream>>>(pe);

  StepParams pd;
  pd.x = henc; pd.sxb = HH; pd.sxt = 1; pd.Fin = 1; pd.K = HH + 1;
  pd.qw0 = qwf; pd.qw1 = qwi; pd.qw2 = qwu; pd.qw3 = qwo;
  pd.Wy = Wy_d; pd.by = by_d;
  pd.Wf = Wf_d; pd.bf = bf_d; pd.Wi = Wi_d; pd.bi = bi_d;
  pd.Wu = Wu_d; pd.bu = bu_d; pd.Wo = Wo_d; pd.bo = bo_d;
  pd.hout = hdec;
  qlstm_step<<<1, NTHREADS, smem, stream>>>(pd);

  qae_final<<<4, 256, 0, stream>>>(hdec, Wt, bt, (float*)d_out);
}